// GravNetOp_1468878815446
// MI455X (gfx1250) — compile-verified
//
#include <hip/hip_runtime.h>
#include <hip/hip_bf16.h>
#include <float.h>

// GravNet: B=4 events x M=4096 points, IN_DIM=128, SPACE=4, PROP=64, K=32.
// Kernel 1: fused linear layers via V_WMMA_F32_16X16X4_F32 (f32 matrix pipe).
// Kernel 2: per-point wave32 exact KNN top-32 (distributed bitonic top-k in
//           registers across lanes) + weighted mean/max aggregation via LDS.

typedef __attribute__((ext_vector_type(2))) float v2f;
typedef __attribute__((ext_vector_type(8))) float v8f;

#define GN_IN_DIM 128
#define GN_SPACE  4
#define GN_PROP   64
#define GN_K      32
#define GN_MSEG   4096

// ---------------------------------------------------------------------------
// Kernel 1: space = x@Ws + bs  [N,4],  prop = x@Wp + bp  [N,64]
// One wave per 16-row tile. K-dim = 128 -> 32 WMMA steps of 16x16x4 f32.
// A-fragment (16x4 f32, ISA table): lanes 0-15 hold (K=0,K=1) of row M=lane,
// lanes 16-31 hold (K=2,K=3) of row M=lane-16. B-fragment mirrors with N=lane.
// ---------------------------------------------------------------------------
__global__ __launch_bounds__(32) void gravnet_linear_wmma(
    const float* __restrict__ x,
    const float* __restrict__ Ws,   // [128][4]
    const float* __restrict__ bs,   // [4]
    const float* __restrict__ Wp,   // [128][64]
    const float* __restrict__ bp,   // [64]
    float* __restrict__ space4,     // [N][4]
    float* __restrict__ prop)       // [N][64]
{
    const int lane = threadIdx.x;
    const bool hi  = lane >= 16;
    const int  ln  = lane & 15;
    const int  r0  = blockIdx.x * 16;

    v8f accP0 = {}, accP1 = {}, accP2 = {}, accP3 = {};
    v8f accS  = {};

#pragma unroll 4
    for (int kk = 0; kk < 32; ++kk) {
        const int k0 = kk * 4;
        // A fragment: one float4 row-chunk per lane, pick lo/hi pair
        const float4 xv = *reinterpret_cast<const float4*>(
            x + (size_t)(r0 + ln) * GN_IN_DIM + k0);
        v2f a;
        a.x = hi ? xv.z : xv.x;
        a.y = hi ? xv.w : xv.y;

        const int ka = k0 + (hi ? 2 : 0);
        const int kb = ka + 1;

        // B fragment for the space GEMM (cols 4..15 zero-padded)
        v2f bS; bS.x = 0.0f; bS.y = 0.0f;
        if (ln < GN_SPACE) {
            bS.x = Ws[ka * GN_SPACE + ln];
            bS.y = Ws[kb * GN_SPACE + ln];
        }
        accS = __builtin_amdgcn_wmma_f32_16x16x4_f32(
            false, a, false, bS, (short)0, accS, false, false);

        // B fragments for the 4 prop N-tiles (64 cols)
        v2f b0, b1, b2, b3;
        b0.x = Wp[ka * GN_PROP + 0  + ln];  b0.y = Wp[kb * GN_PROP + 0  + ln];
        b1.x = Wp[ka * GN_PROP + 16 + ln];  b1.y = Wp[kb * GN_PROP + 16 + ln];
        b2.x = Wp[ka * GN_PROP + 32 + ln];  b2.y = Wp[kb * GN_PROP + 32 + ln];
        b3.x = Wp[ka * GN_PROP + 48 + ln];  b3.y = Wp[kb * GN_PROP + 48 + ln];

        accP0 = __builtin_amdgcn_wmma_f32_16x16x4_f32(
            false, a, false, b0, (short)0, accP0, false, false);
        accP1 = __builtin_amdgcn_wmma_f32_16x16x4_f32(
            false, a, false, b1, (short)0, accP1, false, false);
        accP2 = __builtin_amdgcn_wmma_f32_16x16x4_f32(
            false, a, false, b2, (short)0, accP2, false, false);
        accP3 = __builtin_amdgcn_wmma_f32_16x16x4_f32(
            false, a, false, b3, (short)0, accP3, false, false);
    }

    // C/D layout: VGPR v holds row M = v + (hi?8:0), col N = ln.
    const int rbase = r0 + (hi ? 8 : 0);

    // store prop (+bias)
    {
        const float bp0 = bp[0  + ln];
        const float bp1 = bp[16 + ln];
        const float bp2 = bp[32 + ln];
        const float bp3 = bp[48 + ln];
#pragma unroll
        for (int v = 0; v < 8; ++v) {
            const size_t r = (size_t)(rbase + v) * GN_PROP;
            prop[r + 0  + ln] = accP0[v] + bp0;
            prop[r + 16 + ln] = accP1[v] + bp1;
            prop[r + 32 + ln] = accP2[v] + bp2;
            prop[r + 48 + ln] = accP3[v] + bp3;
        }
    }
    // store space (+bias), only cols 0..3 are real
    if (ln < GN_SPACE) {
        const float bsv = bs[ln];
#pragma unroll
        for (int v = 0; v < 8; ++v) {
            space4[(size_t)(rbase + v) * GN_SPACE + ln] = accS[v] + bsv;
        }
    }
}

// ---------------------------------------------------------------------------
// Cross-lane compare-and-swap for bitonic networks over the wave32.
// up==true  -> the lower lane of the pair keeps the smaller value.
// ---------------------------------------------------------------------------
__device__ __forceinline__ void cas32(float& v, int& i, int j, bool up) {
    const float ov = __shfl_xor(v, j, 32);
    const int   oi = __shfl_xor(i, j, 32);
    const bool lower       = ((threadIdx.x & j) == 0);
    const bool takeSmaller = (lower == up);
    const bool sw = takeSmaller ? (ov < v) : (ov > v);
    v = sw ? ov : v;
    i = sw ? oi : i;
}

// ---------------------------------------------------------------------------
// Kernel 2: one wave32 per point. Exact top-32 smallest d2 within the event,
// held distributed across lanes sorted ascending (lane 31 = current max).
// Then per-lane neighbor gather + exp weight, LDS transpose, mean/max reduce.
// ---------------------------------------------------------------------------
__global__ __launch_bounds__(32) void gravnet_knn_aggregate(
    const float* __restrict__ space4,  // [N][4]
    const float* __restrict__ prop,    // [N][64]
    float* __restrict__ out)           // [N][128]
{
    __shared__ float red[GN_K * 68];   // stride 68 floats: bank-conflict-free

    const int lane = threadIdx.x;
    const int row  = blockIdx.x;
    const int e0   = row & ~(GN_MSEG - 1);   // event base (uniform segments)

    const float4 s = *reinterpret_cast<const float4*>(space4 + (size_t)row * 4);

    float bestV = 3.0e38f;   // ascending across lanes
    int   bestI = e0;

#pragma unroll 1
    for (int t = 0; t < GN_MSEG / 32; ++t) {
        const int col = e0 + t * 32 + lane;
        const float4 c = *reinterpret_cast<const float4*>(space4 + (size_t)col * 4);
        const float dx = s.x - c.x, dy = s.y - c.y;
        const float dz = s.z - c.z, dw = s.w - c.w;
        float cv = dx * dx + dy * dy + dz * dz + dw * dw;

        const float curMax = __shfl(bestV, 31, 32);
        if (__any(cv < curMax)) {
            int ci = col;
            // bitonic sort candidates DESCENDING across lanes
#pragma unroll
            for (int k = 2; k <= 32; k <<= 1) {
                const bool up = (lane & k) != 0;   // inverted -> overall descending
#pragma unroll
                for (int j = k >> 1; j >= 1; j >>= 1) cas32(cv, ci, j, up);
            }
            // best(asc) ++ cand(desc) is bitonic of 64; elementwise min keeps
            // the 32 smallest (itself a bitonic sequence)
            const bool take = cv < bestV;
            bestV = take ? cv : bestV;
            bestI = take ? ci : bestI;
            // cleanup merge back to ascending
#pragma unroll
            for (int j = 16; j >= 1; j >>= 1) cas32(bestV, bestI, j, true);
        }
    }

    // Each lane owns one of the 32 nearest neighbors (self included, d2=0).
    const float w = __expf(-10.0f * fmaxf(bestV, 0.0f));
    const float* pr = prop + (size_t)bestI * GN_PROP;

#pragma unroll
    for (int jv = 0; jv < GN_PROP / 4; ++jv) {
        float4 p = *reinterpret_cast<const float4*>(pr + jv * 4);
        p.x *= w; p.y *= w; p.z *= w; p.w *= w;
        *reinterpret_cast<float4*>(&red[lane * 68 + jv * 4]) = p;
    }
    __syncthreads();

    // 2 features per lane: weighted mean and max over the 32 neighbors
#pragma unroll
    for (int rep = 0; rep < 2; ++rep) {
        const int f = lane + rep * 32;
        float sum = 0.0f, mx = -3.0e38f;
#pragma unroll 8
        for (int n = 0; n < GN_K; ++n) {
            const float v = red[n * 68 + f];
            sum += v;
            mx = fmaxf(mx, v);
        }
        out[(size_t)row * 128 + f]      = sum * (1.0f / (float)GN_K);
        out[(size_t)row * 128 + 64 + f] = mx;
    }
}

// ---------------------------------------------------------------------------
extern "C" void kernel_launch(void* const* d_in, const int* in_sizes, int n_in,
                              void* d_out, int out_size, void* d_ws, size_t ws_size,
                              hipStream_t stream) {
    const float* x  = (const float*)d_in[0];
    const float* Ws = (const float*)d_in[1];
    const float* bs = (const float*)d_in[2];
    const float* Wp = (const float*)d_in[3];
    const float* bp = (const float*)d_in[4];
    // d_in[5] = row_splits (uniform segments of 4096) -- not needed

    float* out = (float*)d_out;
    const int N = in_sizes[0] / GN_IN_DIM;   // 16384

    // workspace: prop [N][64] then space4 [N][4]  (~4.45 MB)
    float* prop   = (float*)d_ws;
    float* space4 = prop + (size_t)N * GN_PROP;

    gravnet_linear_wmma<<<N / 16, 32, 0, stream>>>(x, Ws, bs, Wp, bp, space4, prop);
    gravnet_knn_aggregate<<<N, 32, 0, stream>>>(space4, prop, out);
}